// NUFFTLayer_36172214567404
// MI455X (gfx1250) — compile-verified
//
#include <hip/hip_runtime.h>

typedef float v2f __attribute__((ext_vector_type(2)));
typedef float v4f __attribute__((ext_vector_type(4)));
typedef float v8f __attribute__((ext_vector_type(8)));

#define WAVES 8

__global__ __launch_bounds__(WAVES * 32)
void nufft2_wmma(const float* __restrict__ xr, const float* __restrict__ xi,
                 const float* __restrict__ traj, float* __restrict__ out)
{
    constexpr int B = 16, C = 4, H = 64, W = 64, M = 2048;
    (void)B; (void)H;

    // Per-wave E_y table: 16 m-values x 64 rows, (cos, sin) pairs. 8KB/wave.
    __shared__ float2 eyTab[WAVES][16][64];

    const int lane = threadIdx.x & 31;
    const int wave = threadIdx.x >> 5;
    const int tile = blockIdx.x * WAVES + wave;   // 0..2047 (frame-b m-tile)
    const int b    = tile >> 7;                   // 128 m-tiles per frame
    const int m0   = (tile & 127) << 4;           // first m of this tile
    const int hh   = lane >> 4;                   // wave half (0/1)
    const int mloc = lane & 15;                   // local m / local n index

    // traj[b, m, :] = (ky, kx)
    const float2 kk = ((const float2*)traj)[(size_t)b * M + m0 + mloc];
    const float ky = kk.x, kx = kk.y;

    // ---- build E_y table via unit-circle recurrence (2 sincos per lane) ----
    {
        const int r0 = hh * 32;                    // lanes<16: r 0..31, else 32..63
        float s0, c0, su, cu;
        sincosf(ky * (float)(r0 - 32), &s0, &c0);  // angle ky*gy at start row
        sincosf(ky, &su, &cu);                     // step rotation by ky
        float cc = c0, ss = s0;
        #pragma unroll 4
        for (int r = 0; r < 32; ++r) {
            eyTab[wave][mloc][r0 + r] = make_float2(cc, ss);
            const float t = cc * cu - ss * su;
            ss = ss * cu + cc * su;
            cc = t;
        }
    }
    __syncthreads();

    // ---- E_x A-fragments, 16 K-chunks ----
    // K-slot -> column mapping for chunk k: {2k, 2k+1, 32+2k, 33+2k}
    // (half-0 lanes supply slots {0,1} = cols 2k,2k+1;
    //  half-1 lanes supply slots {2,3} = cols 32+2k,33+2k)
    // => consecutive chunks are 4 contiguous floats per lane: b128 B-loads.
    v2f axc[16], axs[16];   // cos(kx*gx) and sin(kx*gx) fragments
    {
        const int col0 = hh * 32;
        float s0, c0, su, cu;
        sincosf(kx * (float)(col0 - 32), &s0, &c0);   // gx start: -32 or 0
        sincosf(kx, &su, &cu);                        // rotate by kx (col+1)
        const float c2 = cu * cu - su * su, s2 = 2.0f * cu * su;  // rotate by 2*kx
        float ec = c0, es = s0;
        #pragma unroll
        for (int k = 0; k < 16; ++k) {
            const float e1c = ec * cu - es * su;   // col+1
            const float e1s = es * cu + ec * su;
            axc[k].x = ec;  axc[k].y = e1c;
            axs[k].x = es;  axs[k].y = e1s;
            const float t = ec * c2 - es * s2;     // advance 2 columns (next chunk)
            es = es * c2 + ec * s2;
            ec = t;
        }
    }

    // ---- main loop: 16 N-tiles of (row_local, coil); fuse stage-2 E_y reduce ----
    float yre[8], yim[8];
    #pragma unroll
    for (int j = 0; j < 8; ++j) { yre[j] = 0.0f; yim[j] = 0.0f; }

    const int cc_ = mloc & 3;           // coil
    const int rl  = mloc >> 2;          // row sub-index within tile
    const size_t xbase = ((size_t)(b * C + cc_) * H) * W + (size_t)(hh * 32);

    #pragma unroll 1
    for (int t = 0; t < 16; ++t) {
        const int r = t * 4 + rl;
        const float* pr = xr + xbase + (size_t)r * W;
        const float* pi = xi + xbase + (size_t)r * W;

        // Phase 1: 16 x b128 loads (each feeds two K-chunks), all independent.
        v4f brA[8], biA[8];
        #pragma unroll
        for (int k2 = 0; k2 < 8; ++k2) {
            brA[k2] = *(const v4f*)(pr + 4 * k2);   // x_real cols (4 contiguous)
            biA[k2] = *(const v4f*)(pi + 4 * k2);   // x_imag cols
        }

        // Phase 2: dense WMMA chain.
        // Z_re = cos@Xr + sin@Xi ; Z_im = cos@Xi - sin@Xr
        v8f zr = {}; v8f zi = {};
        #pragma unroll
        for (int k2 = 0; k2 < 8; ++k2) {
            const v2f br0 = __builtin_shufflevector(brA[k2], brA[k2], 0, 1);
            const v2f br1 = __builtin_shufflevector(brA[k2], brA[k2], 2, 3);
            const v2f bi0 = __builtin_shufflevector(biA[k2], biA[k2], 0, 1);
            const v2f bi1 = __builtin_shufflevector(biA[k2], biA[k2], 2, 3);
            const int k0 = 2 * k2, k1 = 2 * k2 + 1;

            zr = __builtin_amdgcn_wmma_f32_16x16x4_f32(false, axc[k0], false, br0,
                                                       (short)0, zr, false, false);
            zi = __builtin_amdgcn_wmma_f32_16x16x4_f32(false, axc[k0], false, bi0,
                                                       (short)0, zi, false, false);
            zr = __builtin_amdgcn_wmma_f32_16x16x4_f32(false, axs[k0], false, bi0,
                                                       (short)0, zr, false, false);
            const v2f nbr0 = -br0;  // f32 WMMA has no A/B NEG modifier
            zi = __builtin_amdgcn_wmma_f32_16x16x4_f32(false, axs[k0], false, nbr0,
                                                       (short)0, zi, false, false);

            zr = __builtin_amdgcn_wmma_f32_16x16x4_f32(false, axc[k1], false, br1,
                                                       (short)0, zr, false, false);
            zi = __builtin_amdgcn_wmma_f32_16x16x4_f32(false, axc[k1], false, bi1,
                                                       (short)0, zi, false, false);
            zr = __builtin_amdgcn_wmma_f32_16x16x4_f32(false, axs[k1], false, bi1,
                                                       (short)0, zr, false, false);
            const v2f nbr1 = -br1;
            zi = __builtin_amdgcn_wmma_f32_16x16x4_f32(false, axs[k1], false, nbr1,
                                                       (short)0, zi, false, false);
        }

        // stage 2: y[m,c] += E_y[m,r] * Z[m,r,c]  (E_y = (cy, -sy))
        #pragma unroll
        for (int j = 0; j < 8; ++j) {
            const float2 ey = eyTab[wave][j + hh * 8][r];
            yre[j] = fmaf(ey.x, zr[j], yre[j]);
            yre[j] = fmaf(ey.y, zi[j], yre[j]);
            yim[j] = fmaf(ey.x, zi[j], yim[j]);
            yim[j] = fmaf(-ey.y, zr[j], yim[j]);
        }
    }

    // ---- reduce over the 4 row-sub lanes (xor 4, 8) and store [B,C,M,2] ----
    #pragma unroll
    for (int j = 0; j < 8; ++j) {
        float a = yre[j], c = yim[j];
        a += __shfl_xor(a, 4, 32);
        a += __shfl_xor(a, 8, 32);
        c += __shfl_xor(c, 4, 32);
        c += __shfl_xor(c, 8, 32);
        if ((lane & 12) == 0) {
            const int m  = m0 + j + hh * 8;
            const int ch = lane & 3;
            float2* op = (float2*)out + ((size_t)(b * C + ch) * M + m);
            *op = make_float2(a, c);
        }
    }
}

extern "C" void kernel_launch(void* const* d_in, const int* in_sizes, int n_in,
                              void* d_out, int out_size, void* d_ws, size_t ws_size,
                              hipStream_t stream) {
    (void)in_sizes; (void)n_in; (void)out_size; (void)d_ws; (void)ws_size;
    const float* xr   = (const float*)d_in[0];
    const float* xi   = (const float*)d_in[1];
    const float* traj = (const float*)d_in[2];
    float* out = (float*)d_out;

    // 16 frames * 128 m-tiles = 2048 waves; 8 waves/block -> 256 blocks
    dim3 grid(256), block(WAVES * 32);
    hipLaunchKernelGGL(nufft2_wmma, grid, block, 0, stream, xr, xi, traj, out);
}